// SequencePredictor_58583353917999
// MI455X (gfx1250) — compile-verified
//
#include <hip/hip_runtime.h>
#include <stdint.h>

// ---------------------------------------------------------------------------
// h_t = sigmoid(x_t*Wx + h_{t-1}*Wh + b), out = sigmoid(h_T).
// tanh-space reformulation:
//   t_i = tanh(u_i),  h_i = 0.5*t_i + 0.5,  t_0 = -1
//   u_i = (0.25*Wh)*t_{i-1} + [0.5*(Wx*x_i + b) + 0.25*Wh]
// Dependent chain per timestep: one v_fma_f32 + one v_tanh_f32.
// Data movement: async global->LDS double buffering (ASYNCcnt) on gfx1250.
// ---------------------------------------------------------------------------

#define T_LEN      4096
#define BLOCK      64            // rows per block = 2 wave32s
#define CH         32            // timesteps staged per chunk
#define LDS_STRIDE (CH + 4)      // 36 dwords = 144B: 16B-aligned, conflict-light

#if defined(__gfx1250__) &&                                            \
    __has_builtin(__builtin_amdgcn_global_load_async_to_lds_b128) &&   \
    __has_builtin(__builtin_amdgcn_s_wait_asynccnt)
#define USE_ASYNC 1
#else
#define USE_ASYNC 0
#endif

typedef int v4i __attribute__((ext_vector_type(4)));
typedef __attribute__((address_space(1))) v4i GV4;  // global int4
typedef __attribute__((address_space(3))) v4i LV4;  // LDS int4

__device__ __forceinline__ float fast_tanh(float v) {
#if defined(__gfx1250__) && __has_builtin(__builtin_amdgcn_tanhf)
  return __builtin_amdgcn_tanhf(v);
#elif defined(__gfx1250__) && __has_builtin(__builtin_amdgcn_tanh_f32)
  return __builtin_amdgcn_tanh_f32(v);
#else
  return tanhf(v);
#endif
}

__global__ __launch_bounds__(BLOCK) void rnn_scan_kernel(
    const float* __restrict__ x, const float* __restrict__ Wx,
    const float* __restrict__ Wh, const float* __restrict__ b,
    float* __restrict__ out) {
#if USE_ASYNC
  __shared__ __align__(16) float lds[2][BLOCK * LDS_STRIDE];
#endif
  const int row = blockIdx.x * BLOCK + threadIdx.x;

  const float wx = Wx[0];
  const float wh = Wh[0];
  const float bb = b[0];

  const float c  = 0.25f * wh;               // chain multiplier
  const float s0 = 0.5f * wx;                // off-chain scale
  const float s1 = 0.5f * bb + 0.25f * wh;   // off-chain bias

  const float* xrow = x + (size_t)row * T_LEN;
  float t = -1.0f;                           // t_0 = -1  <=>  h_0 = 0

#if USE_ASYNC
  float* buf0 = &lds[0][threadIdx.x * LDS_STRIDE];
  float* buf1 = &lds[1][threadIdx.x * LDS_STRIDE];

  auto issue = [&](float* dstLds, int tBase) {
    const float* src = xrow + tBase;
#pragma unroll
    for (int q = 0; q < CH / 4; ++q) {
      __builtin_amdgcn_global_load_async_to_lds_b128(
          (GV4*)(uintptr_t)(src + q * 4),
          (LV4*)(uint32_t)(uintptr_t)(dstLds + q * 4),
          /*offset=*/0, /*cpol=*/0);
    }
  };

  issue(buf0, 0);
  const int NCH = T_LEN / CH;
  for (int k = 0; k < NCH; ++k) {
    float* cur = (k & 1) ? buf1 : buf0;
    float* nxt = (k & 1) ? buf0 : buf1;
    if (k + 1 < NCH) {
      issue(nxt, (k + 1) * CH);
      // chunk k's CH/4 ops complete; chunk k+1's CH/4 still in flight
      __builtin_amdgcn_s_wait_asynccnt(CH / 4);
    } else {
      __builtin_amdgcn_s_wait_asynccnt(0);
    }
#pragma unroll
    for (int i = 0; i < CH; i += 4) {
      float4 v = *(const float4*)(cur + i);           // ds_load_b128
      float a0 = fmaf(s0, v.x, s1);                   // off-chain
      float a1 = fmaf(s0, v.y, s1);
      float a2 = fmaf(s0, v.z, s1);
      float a3 = fmaf(s0, v.w, s1);
      t = fast_tanh(fmaf(c, t, a0));                  // chain: fma + tanh
      t = fast_tanh(fmaf(c, t, a1));
      t = fast_tanh(fmaf(c, t, a2));
      t = fast_tanh(fmaf(c, t, a3));
    }
  }
#else
  const float4* xv = (const float4*)xrow;
#pragma unroll 4
  for (int i = 0; i < T_LEN / 4; ++i) {
    float4 v = xv[i];
    float a0 = fmaf(s0, v.x, s1);
    float a1 = fmaf(s0, v.y, s1);
    float a2 = fmaf(s0, v.z, s1);
    float a3 = fmaf(s0, v.w, s1);
    t = fast_tanh(fmaf(c, t, a0));
    t = fast_tanh(fmaf(c, t, a1));
    t = fast_tanh(fmaf(c, t, a2));
    t = fast_tanh(fmaf(c, t, a3));
  }
#endif

  const float h = fmaf(0.5f, t, 0.5f);               // h_T
  // out = sigmoid(h_T) = 0.5*tanh(0.5*h_T) + 0.5
  out[row] = fmaf(0.5f, fast_tanh(0.5f * h), 0.5f);
}

extern "C" void kernel_launch(void* const* d_in, const int* in_sizes, int n_in,
                              void* d_out, int out_size, void* d_ws, size_t ws_size,
                              hipStream_t stream) {
  (void)n_in; (void)d_ws; (void)ws_size; (void)out_size;
  const float* x  = (const float*)d_in[0];
  const float* Wx = (const float*)d_in[1];
  const float* Wh = (const float*)d_in[2];
  const float* bb = (const float*)d_in[3];
  float* out = (float*)d_out;

  const int batch = in_sizes[0] / T_LEN;   // 16384
  const int grid  = batch / BLOCK;         // 256 blocks of 2 waves

  rnn_scan_kernel<<<grid, BLOCK, 0, stream>>>(x, Wx, Wh, bb, out);
}